// SpatioTemporalGNN_56822417326343
// MI455X (gfx1250) — compile-verified
//
#include <hip/hip_runtime.h>
#include <hip/hip_bf16.h>

typedef float v2f __attribute__((ext_vector_type(2)));
typedef float v8f __attribute__((ext_vector_type(8)));

#define HD 64  // hidden dim

// ---------------------------------------------------------------------------
// Degree / dinv
// ---------------------------------------------------------------------------
__global__ void deg_init(float* deg, int n) {
    int i = blockIdx.x * blockDim.x + threadIdx.x;
    if (i < n) deg[i] = 1.0f;  // self-loop contributes 1
}

__global__ void deg_accum(const int* __restrict__ col, float* deg, int E) {
    int e = blockIdx.x * blockDim.x + threadIdx.x;
    if (e < E) atomicAdd(&deg[col[e]], 1.0f);
}

__global__ void dinv_fin(float* deg, int n) {
    int i = blockIdx.x * blockDim.x + threadIdx.x;
    if (i < n) {
        float d = deg[i];
        deg[i] = d > 0.0f ? rsqrtf(d) : 0.0f;
    }
}

// ---------------------------------------------------------------------------
// Layer-1 dense: xw[n,c] = sum_{k<5} x[n,k] * w1[k,c]   (K too small for WMMA)
// ---------------------------------------------------------------------------
__global__ void lin_in5(const float* __restrict__ x, const float* __restrict__ w,
                        float* __restrict__ xw, int total /* = N*64 */) {
    int i = blockIdx.x * blockDim.x + threadIdx.x;
    if (i >= total) return;
    int n = i >> 6, c = i & 63;
    const float* xr = x + (long long)n * 5;
    float acc = 0.0f;
#pragma unroll
    for (int k = 0; k < 5; ++k) acc += xr[k] * w[k * HD + c];
    xw[i] = acc;
}

// ---------------------------------------------------------------------------
// WMMA GEMM: Out[M,64] = A[M,64] @ W[64,64]   (fp32, V_WMMA_F32_16X16X4_F32)
// One wave32 per 16x16 output tile; 16 WMMAs over K=64.
// A frag (16x4 f32): lanes 0-15 -> M=lane, VGPR0/1 = K+0/K+1;
//                    lanes 16-31 -> M=lane-16, VGPR0/1 = K+2/K+3.
// C/D (16x16 f32): VGPR v: lanes 0-15 -> M=v, lanes 16-31 -> M=v+8.
// ---------------------------------------------------------------------------
__global__ void gemm64_wmma(const float* __restrict__ A, const float* __restrict__ W,
                            float* __restrict__ Out, int M) {
    int wv   = (blockIdx.x * blockDim.x + threadIdx.x) >> 5;
    int lane = threadIdx.x & 31;
    int totalTiles = (M >> 4) * 4;           // 4 N-tiles of 16 cols
    if (wv >= totalTiles) return;            // wave-uniform exit (EXEC all-1s for WMMA)
    int mt = wv >> 2, nt = wv & 3;
    int m0 = mt << 4, n0 = nt << 4;
    int half = lane >> 4, l16 = lane & 15;

    const float* arow  = A + (long long)(m0 + l16) * HD + 2 * half;
    const float* bbase = W + (2 * half) * HD + n0 + l16;

    v8f acc = {};
#pragma unroll
    for (int k = 0; k < HD; k += 4) {
        v2f a, b;
        a.x = arow[k];
        a.y = arow[k + 1];
        b.x = bbase[k * HD];          // W[k + 2*half    ][n0+l16]
        b.y = bbase[k * HD + HD];     // W[k + 2*half + 1][n0+l16]
        acc = __builtin_amdgcn_wmma_f32_16x16x4_f32(
            /*neg_a=*/false, a, /*neg_b=*/false, b,
            /*c_mod=*/(short)0, acc, /*reuse_a=*/false, /*reuse_b=*/false);
    }
#pragma unroll
    for (int v = 0; v < 8; ++v)
        Out[(long long)(m0 + v + 8 * half) * HD + n0 + l16] = acc[v];
}

// ---------------------------------------------------------------------------
// agg[n,c] = dinv[n]^2 * xw[n,c] + bias[c]   (self-loop term + post-agg bias;
// also initializes agg buffer)
// ---------------------------------------------------------------------------
__global__ void init_agg(const float* __restrict__ xw, const float* __restrict__ dinv,
                         const float* __restrict__ bias, float* __restrict__ agg,
                         int total) {
    int i = blockIdx.x * blockDim.x + threadIdx.x;
    if (i >= total) return;
    int n = i >> 6, c = i & 63;
    float d = dinv[n];
    agg[i] = d * d * xw[i] + bias[c];
}

// ---------------------------------------------------------------------------
// Edge scatter: agg[col] += dinv[row]*dinv[col] * xw[row]; 16 threads/edge,
// float4 per thread, 4 f32 atomics (L2-side RMW).
// ---------------------------------------------------------------------------
__global__ void edge_scatter(const float* __restrict__ xw, const int* __restrict__ rows,
                             const int* __restrict__ cols, const float* __restrict__ dinv,
                             float* __restrict__ agg, int E) {
    long long i = (long long)blockIdx.x * blockDim.x + threadIdx.x;
    long long total = (long long)E * 16;
    if (i >= total) return;
    int e  = (int)(i >> 4);
    int c4 = ((int)i & 15) << 2;
    int r = rows[e], c = cols[e];
    float nrm = dinv[r] * dinv[c];
    const float4 v = *reinterpret_cast<const float4*>(xw + (long long)r * HD + c4);
    float* dst = agg + (long long)c * HD + c4;
    atomicAdd(dst + 0, nrm * v.x);
    atomicAdd(dst + 1, nrm * v.y);
    atomicAdd(dst + 2, nrm * v.z);
    atomicAdd(dst + 3, nrm * v.w);
}

// ---------------------------------------------------------------------------
// BatchNorm: stats -> finalize -> apply(+ReLU)
// ---------------------------------------------------------------------------
__global__ void zero_stats(float* stats) {
    if (threadIdx.x < 128) stats[threadIdx.x] = 0.0f;
}

// grid stride is a multiple of 64 so each thread's channel is fixed ->
// register-private partials, then LDS atomics, then 128 global atomics/block.
__global__ void bn_stats(const float* __restrict__ h, float* __restrict__ stats,
                         long long total) {
    __shared__ float ss[128];
    if (threadIdx.x < 128) ss[threadIdx.x] = 0.0f;
    __syncthreads();
    long long start  = (long long)blockIdx.x * blockDim.x + threadIdx.x;
    long long stride = (long long)gridDim.x * blockDim.x;   // 512*256, %64==0
    int c = (int)(start & 63);
    float s = 0.0f, q = 0.0f;
    for (long long i = start; i < total; i += stride) {
        float v = h[i];
        s += v;
        q += v * v;
    }
    atomicAdd(&ss[c], s);
    atomicAdd(&ss[64 + c], q);
    __syncthreads();
    if (threadIdx.x < 128) atomicAdd(&stats[threadIdx.x], ss[threadIdx.x]);
}

// stats[c] <- scale = g*rsqrt(var+eps); stats[64+c] <- shift = beta - mean*scale
__global__ void bn_finalize(float* stats, const float* __restrict__ g,
                            const float* __restrict__ be, float invN) {
    int c = threadIdx.x;
    if (c >= HD) return;
    float mean = stats[c] * invN;
    float var  = stats[64 + c] * invN - mean * mean;
    float scl  = g[c] * rsqrtf(var + 1e-5f);
    stats[c]      = scl;
    stats[64 + c] = be[c] - mean * scl;
}

__global__ void bn_apply_relu(float* __restrict__ h, const float* __restrict__ stats,
                              int total) {
    int i = blockIdx.x * blockDim.x + threadIdx.x;
    if (i >= total) return;
    int c = i & 63;
    float v = h[i] * stats[c] + stats[64 + c];
    h[i] = fmaxf(v, 0.0f);
}

// ---------------------------------------------------------------------------
// Final FC: out[n, 0..3] = h[n,:] @ wfc + bfc   (N=4, skinny -> VALU)
// ---------------------------------------------------------------------------
__global__ void fc_out(const float* __restrict__ h, const float* __restrict__ wfc,
                       const float* __restrict__ bfc, float* __restrict__ out, int Nn) {
    int n = blockIdx.x * blockDim.x + threadIdx.x;
    if (n >= Nn) return;
    float a0 = bfc[0], a1 = bfc[1], a2 = bfc[2], a3 = bfc[3];
    const float* hr = h + (long long)n * HD;
#pragma unroll
    for (int k = 0; k < HD; ++k) {
        float v = hr[k];
        a0 += v * wfc[k * 4 + 0];
        a1 += v * wfc[k * 4 + 1];
        a2 += v * wfc[k * 4 + 2];
        a3 += v * wfc[k * 4 + 3];
    }
    float* o = out + (long long)n * 4;
    o[0] = a0; o[1] = a1; o[2] = a2; o[3] = a3;
}

// ---------------------------------------------------------------------------
// Host orchestration
// ---------------------------------------------------------------------------
static inline int cdiv(long long a, int b) { return (int)((a + b - 1) / b); }

extern "C" void kernel_launch(void* const* d_in, const int* in_sizes, int n_in,
                              void* d_out, int out_size, void* d_ws, size_t ws_size,
                              hipStream_t stream) {
    const float* x       = (const float*)d_in[0];
    const int*   e_sp    = (const int*)d_in[1];   // [2, E] flat: rows then cols
    const int*   e_tm    = (const int*)d_in[2];
    const float* W[4]    = {(const float*)d_in[3],  (const float*)d_in[7],
                            (const float*)d_in[11], (const float*)d_in[15]};
    const float* B[4]    = {(const float*)d_in[4],  (const float*)d_in[8],
                            (const float*)d_in[12], (const float*)d_in[16]};
    const float* G[4]    = {(const float*)d_in[5],  (const float*)d_in[9],
                            (const float*)d_in[13], (const float*)d_in[17]};
    const float* BE[4]   = {(const float*)d_in[6],  (const float*)d_in[10],
                            (const float*)d_in[14], (const float*)d_in[18]};
    const float* wfc     = (const float*)d_in[19];
    const float* bfc     = (const float*)d_in[20];

    const int N = in_sizes[0] / 5;          // 100000 nodes
    const int E = in_sizes[1] / 2;          // 1.6M edges per graph
    const long long NF = (long long)N * HD; // 6.4M activation elements

    float* ws     = (float*)d_ws;
    float* bufA   = ws;                     // xw scratch
    float* bufB   = ws + NF;
    float* bufC   = ws + 2 * NF;
    float* dinvS  = ws + 3 * NF;
    float* dinvT  = dinvS + N;
    float* stats  = dinvT + N;              // 128 floats

    const int T = 256;
    const int totalNF = (int)NF;

    // --- dinv for both graphs (recomputed each call: deterministic) ---
    deg_init<<<cdiv(N, T), T, 0, stream>>>(dinvS, N);
    deg_init<<<cdiv(N, T), T, 0, stream>>>(dinvT, N);
    deg_accum<<<cdiv(E, T), T, 0, stream>>>(e_sp + E, dinvS, E);  // col half
    deg_accum<<<cdiv(E, T), T, 0, stream>>>(e_tm + E, dinvT, E);
    dinv_fin<<<cdiv(N, T), T, 0, stream>>>(dinvS, N);
    dinv_fin<<<cdiv(N, T), T, 0, stream>>>(dinvT, N);

    const int gemmBlocks    = cdiv((long long)(N >> 4) * 4 * 32, T); // wave per tile
    const int scatterBlocks = cdiv((long long)E * 16, T);
    const float invN = 1.0f / (float)N;

    // layer buffers: xw always in bufA; agg ping-pongs bufB <-> bufC
    float* agg[4] = {bufB, bufC, bufB, bufC};
    const float* hin[4];
    const int* rows[4] = {e_sp, e_sp, e_tm, e_tm};
    const int* cols[4] = {e_sp + E, e_sp + E, e_tm + E, e_tm + E};
    float* dv[4] = {dinvS, dinvS, dinvT, dinvT};

    for (int L = 0; L < 4; ++L) {
        // 1) xw = h @ W  (WMMA for K=64 layers; VALU for the K=5 input layer)
        if (L == 0) {
            lin_in5<<<cdiv(NF, T), T, 0, stream>>>(x, W[0], bufA, totalNF);
        } else {
            gemm64_wmma<<<gemmBlocks, T, 0, stream>>>(agg[L - 1], W[L], bufA, N);
        }
        // 2) agg = dinv^2 * xw + bias (init)
        init_agg<<<cdiv(NF, T), T, 0, stream>>>(bufA, dv[L], B[L], agg[L], totalNF);
        // 3) edge scatter with atomics
        edge_scatter<<<scatterBlocks, T, 0, stream>>>(bufA, rows[L], cols[L], dv[L],
                                                      agg[L], E);
        // 4-6) BatchNorm (training-mode) + ReLU, applied in place
        zero_stats<<<1, 128, 0, stream>>>(stats);
        bn_stats<<<512, T, 0, stream>>>(agg[L], stats, NF);
        bn_finalize<<<1, 64, 0, stream>>>(stats, G[L], BE[L], invN);
        bn_apply_relu<<<cdiv(NF, T), T, 0, stream>>>(agg[L], stats, totalNF);
    }

    // final FC -> d_out [N,4]
    fc_out<<<cdiv(N, T), T, 0, stream>>>(agg[3], wfc, bfc, (float*)d_out, N);
}